// BaselineDasymetric_26147760898484
// MI455X (gfx1250) — compile-verified
//
#include <hip/hip_runtime.h>

// ---- problem constants (fixed shapes from reference setup) ----
#define LAMBDA_L 0.01f
#define LAMBDA_S 0.01f
#define EPSF     1e-8f

constexpr int NB     = 8;             // batches
constexpr int HW     = 1024 * 1024;   // pixels per batch
constexpr int NADMIN = 512;           // admin bins
constexpr int BPB    = 64;            // blocks per batch (= K partials per bin)
constexpr int TPB    = 256;           // threads per block (8 waves, wave32)
constexpr int PPB    = HW / BPB;      // pixels per block = 16384
constexpr int ITERS  = PPB / (TPB * 4); // float4 per thread iterations = 16

typedef __attribute__((ext_vector_type(2))) float v2f;
typedef __attribute__((ext_vector_type(8))) float v8f;

// ---------------------------------------------------------------------------
// Pass 1: per-block LDS-privatized segment sum of score into 512 bins,
// block partials written to workspace (deterministic layout, no global atomics).
// partial[(b*BPB + blk)*NADMIN + a]
// ---------------------------------------------------------------------------
__global__ __launch_bounds__(TPB)
void k_partial(const float* __restrict__ lights,
               const float* __restrict__ settle,
               const int*   __restrict__ admin,
               float*       __restrict__ partial) {
  __shared__ float hist[NADMIN];
  const int t = threadIdx.x;
  const int b = blockIdx.y;
  for (int i = t; i < NADMIN; i += TPB) hist[i] = 0.0f;
  __syncthreads();

  const long base = (long)b * HW + (long)blockIdx.x * PPB;
  const float4* __restrict__ L = (const float4*)(lights + base);
  const float4* __restrict__ S = (const float4*)(settle + base);
  const int4*   __restrict__ A = (const int4*)(admin + base);

  for (int i = 0; i < ITERS; ++i) {
    const int idx = i * TPB + t;
    if (i + 1 < ITERS) {
      // gfx1250 global_prefetch_b8 — pull next tile toward WGP while we bin this one
      __builtin_prefetch((const void*)(L + idx + TPB), 0, 3);
      __builtin_prefetch((const void*)(S + idx + TPB), 0, 3);
      __builtin_prefetch((const void*)(A + idx + TPB), 0, 3);
    }
    const float4 l = L[idx];
    const float4 s = S[idx];
    const int4   a = A[idx];
    atomicAdd(&hist[a.x], (l.x + LAMBDA_L) * (s.x + LAMBDA_S)); // ds_add_f32
    atomicAdd(&hist[a.y], (l.y + LAMBDA_L) * (s.y + LAMBDA_S));
    atomicAdd(&hist[a.z], (l.z + LAMBDA_L) * (s.z + LAMBDA_S));
    atomicAdd(&hist[a.w], (l.w + LAMBDA_L) * (s.w + LAMBDA_S));
  }
  __syncthreads();

  float* __restrict__ p = partial + (long)(b * BPB + blockIdx.x) * NADMIN;
  for (int i = t; i < NADMIN; i += TPB) p[i] = hist[i];
}

// ---------------------------------------------------------------------------
// Pass 2: reduce the 64 block-partials per bin with V_WMMA_F32_16X16X4_F32.
// sums[m] = sum_k partial[k][m] == (A x ones) row sums. One wave handles 16
// bins (M=16), K looped 64/4. A layout (ISA 7.12.2, 32-bit A 16x4): lanes
// 0-15 carry K=0 (v0) / K=1 (v1), lanes 16-31 carry K=2 / K=3, M = lane%16.
// B = ones -> its lane layout is irrelevant. D: lane n, VGPR v = D[v + 8*(n>=16), n];
// all N columns identical, so lane 0 holds rowsums 0..7, lane 16 holds 8..15.
// Then factor = census / (sum + eps). Total bins = NB*NADMIN = 4096 -> 256 waves.
// ---------------------------------------------------------------------------
__global__ __launch_bounds__(128)
void k_reduce_wmma(const float* __restrict__ partial,
                   const float* __restrict__ census,
                   float*       __restrict__ factor) {
  const int wave = (blockIdx.x * blockDim.x + threadIdx.x) >> 5;
  const int lane = threadIdx.x & 31;
  const int half = lane >> 4;     // 0: K pair {0,1}, 1: K pair {2,3}
  const int row  = lane & 15;     // M within the 16-bin tile

  const int gbase = wave * 16;          // global bin base [0, 4096)
  const int b     = gbase / NADMIN;     // 512 % 16 == 0 -> tile stays in one batch
  const int a0    = gbase % NADMIN;

  const float* __restrict__ P = partial + (long)b * BPB * NADMIN + a0 + row;

  v8f c = {};
  v2f one; one.x = 1.0f; one.y = 1.0f;

  for (int k = 0; k < BPB; k += 4) {
    const int kk = k + 2 * half;
    v2f av;
    av.x = P[(long)kk * NADMIN];        // A[M=row, K=kk]
    av.y = P[(long)(kk + 1) * NADMIN];  // A[M=row, K=kk+1]
    // D = A(16x4) * ones(4x16) + C  -> chained row-sum accumulation
    c = __builtin_amdgcn_wmma_f32_16x16x4_f32(
        /*neg_a=*/false, av, /*neg_b=*/false, one,
        /*c_mod=*/(short)0, c, /*reuse_a=*/false, /*reuse_b=*/false);
  }

  if (row == 0) {                        // lanes 0 and 16 hold the 16 row sums
    const int abase = a0 + half * 8;
    #pragma unroll
    for (int v = 0; v < 8; ++v) {
      const int a = abase + v;
      factor[b * NADMIN + a] = census[a] / (c[v] + EPSF);
    }
  }
}

// ---------------------------------------------------------------------------
// Pass 3: out = score * factor[b, admin]. Inputs are L2-hot from pass 1
// (128 MB working set < 192 MB L2). Factor table staged in 2 KB of LDS.
// ---------------------------------------------------------------------------
__global__ __launch_bounds__(TPB)
void k_apply(const float* __restrict__ lights,
             const float* __restrict__ settle,
             const int*   __restrict__ admin,
             const float* __restrict__ factor,
             float*       __restrict__ out) {
  __shared__ float f[NADMIN];
  const int t = threadIdx.x;
  const int b = blockIdx.y;
  for (int i = t; i < NADMIN; i += TPB) f[i] = factor[b * NADMIN + i];
  __syncthreads();

  const long base = (long)b * HW + (long)blockIdx.x * PPB;
  const float4* __restrict__ L = (const float4*)(lights + base);
  const float4* __restrict__ S = (const float4*)(settle + base);
  const int4*   __restrict__ A = (const int4*)(admin + base);
  float4*       __restrict__ O = (float4*)(out + base);

  for (int i = 0; i < ITERS; ++i) {
    const int idx = i * TPB + t;
    if (i + 1 < ITERS) {
      __builtin_prefetch((const void*)(L + idx + TPB), 0, 3);
      __builtin_prefetch((const void*)(S + idx + TPB), 0, 3);
      __builtin_prefetch((const void*)(A + idx + TPB), 0, 3);
    }
    const float4 l = L[idx];
    const float4 s = S[idx];
    const int4   a = A[idx];
    float4 o;
    o.x = (l.x + LAMBDA_L) * (s.x + LAMBDA_S) * f[a.x];
    o.y = (l.y + LAMBDA_L) * (s.y + LAMBDA_S) * f[a.y];
    o.z = (l.z + LAMBDA_L) * (s.z + LAMBDA_S) * f[a.z];
    o.w = (l.w + LAMBDA_L) * (s.w + LAMBDA_S) * f[a.w];
    O[idx] = o;
  }
}

// ---------------------------------------------------------------------------
extern "C" void kernel_launch(void* const* d_in, const int* in_sizes, int n_in,
                              void* d_out, int out_size, void* d_ws, size_t ws_size,
                              hipStream_t stream) {
  const float* lights = (const float*)d_in[0];
  const float* settle = (const float*)d_in[1];
  const int*   admin  = (const int*)  d_in[2];
  const float* census = (const float*)d_in[3];
  float*       out    = (float*)d_out;

  // workspace layout: [ partials: NB*BPB*NADMIN f32 ][ factors: NB*NADMIN f32 ]
  float* partial = (float*)d_ws;
  float* factor  = partial + (size_t)NB * BPB * NADMIN;

  dim3 gridStream(BPB, NB);

  k_partial<<<gridStream, TPB, 0, stream>>>(lights, settle, admin, partial);

  // 4096 bins / 16 bins-per-wave = 256 waves; 128 threads (4 waves) per block
  k_reduce_wmma<<<(NB * NADMIN / 16) * 32 / 128, 128, 0, stream>>>(partial, census, factor);

  k_apply<<<gridStream, TPB, 0, stream>>>(lights, settle, admin, factor, out);
}